// DigitConvolutionalModel_13537736917524
// MI455X (gfx1250) — compile-verified
//
#include <hip/hip_runtime.h>

// ---------------------------------------------------------------------------
// Fused: 3x3 valid conv (cross-correlation) + ReLU + [B,676]x[676,10] GEMM + b
// MI455X / gfx1250: wave32, V_WMMA_F32_16X16X4_F32 (fp32 in, fp32 acc).
// Memory-bound (~104MB traffic, ~4.5us floor @ 23.3TB/s) -> one fused pass:
//   x read once via CDNA5 async direct-to-LDS (GLOBAL_LOAD_ASYNC_TO_LDS_B128),
//   conv tile in LDS, fp32 WMMA for the 676->10 GEMM, K split over 4 waves.
// ---------------------------------------------------------------------------

#define IMGDIM 28
#define OUTD   26                 // 28 - 3 + 1
#define FEAT   (OUTD * OUTD)      // 676
#define HSTR   680                // padded K stride (even -> 8B aligned pairs)
#define TPB    128                // 4 waves
#define IPB    16                 // images per block (one WMMA M-tile)

typedef float v2f __attribute__((ext_vector_type(2)));
typedef float v8f __attribute__((ext_vector_type(8)));

// Async-to-LDS builtin pointee type (per hipcc diagnostic: vector_size(16) int)
typedef int v4i __attribute__((vector_size(16)));
typedef __attribute__((address_space(1))) v4i g_v4i;   // global
typedef __attribute__((address_space(3))) v4i l_v4i;   // LDS

#if __has_builtin(__builtin_amdgcn_global_load_async_to_lds_b128)
#define HAVE_ASYNC_LDS 1
#else
#define HAVE_ASYNC_LDS 0
#endif

__global__ __launch_bounds__(TPB)
void fused_conv_relu_gemm(const float* __restrict__ x,
                          const float* __restrict__ cw,
                          const float* __restrict__ W,
                          const float* __restrict__ bias,
                          float* __restrict__ out)
{
    // LDS plan (floats):
    //  xs : 16*784 = 12544  raw images; later reused as Wt[16][HSTR] (10880)
    //  hs : 16*680 = 10880  relu(conv) activation tile (WMMA A operand)
    //  red: 4*256  = 1024   per-wave WMMA partials for the K-split reduction
    __shared__ float xs[IPB * 784];
    __shared__ float hs[IPB * HSTR];
    __shared__ float red[4 * 256];

    const int tid     = threadIdx.x;
    const int lane    = tid & 31;
    const int wv      = __builtin_amdgcn_readfirstlane(tid >> 5);  // scalar wave id
    const int imgbase = blockIdx.x * IPB;

    // ---- Phase 1: stage 16 images into LDS --------------------------------
    {
        const float4* gx = (const float4*)(x + (size_t)imgbase * (IMGDIM * IMGDIM));
        float4* sx = (float4*)xs;
        const int n4 = IPB * 784 / 4;            // 3136 float4
#if HAVE_ASYNC_LDS
        // CDNA5 async copy: global -> LDS directly, no VGPR round-trip.
        // AS3 pointer value = LDS byte offset = low 32 bits of the flat address.
        for (int i = tid; i < n4; i += TPB) {
            g_v4i* g = (g_v4i*)(unsigned long long)(gx + i);
            l_v4i* l = (l_v4i*)(unsigned int)(unsigned long long)(sx + i);
            __builtin_amdgcn_global_load_async_to_lds_b128(g, l, 0, 0);
        }
#if __has_builtin(__builtin_amdgcn_s_wait_asynccnt)
        __builtin_amdgcn_s_wait_asynccnt(0);
#else
        asm volatile("s_wait_asynccnt 0x0" ::: "memory");
#endif
#else
        for (int i = tid; i < n4; i += TPB) sx[i] = gx[i];
#endif
    }
    __syncthreads();

    // ---- Phase 2: 3x3 conv + ReLU, one (img, out-row) task per iteration ---
    {
        const float w00 = cw[0], w01 = cw[1], w02 = cw[2];
        const float w10 = cw[3], w11 = cw[4], w12 = cw[5];
        const float w20 = cw[6], w21 = cw[7], w22 = cw[8];

        for (int task = tid; task < IPB * OUTD; task += TPB) {
            const int img = task / OUTD;
            const int row = task - img * OUTD;
            const float* r0 = xs + img * 784 + row * IMGDIM;
            const float* r1 = r0 + IMGDIM;
            const float* r2 = r1 + IMGDIM;

            float a0[IMGDIM], a1[IMGDIM], a2[IMGDIM];
            #pragma unroll
            for (int i = 0; i < 7; ++i) {        // 28 floats = 7 x float4, 16B aligned
                ((float4*)a0)[i] = ((const float4*)r0)[i];
                ((float4*)a1)[i] = ((const float4*)r1)[i];
                ((float4*)a2)[i] = ((const float4*)r2)[i];
            }

            float* hrow = hs + img * HSTR + row * OUTD;
            #pragma unroll
            for (int c = 0; c < OUTD; ++c) {
                float s = a0[c]   * w00 + a0[c+1] * w01 + a0[c+2] * w02
                        + a1[c]   * w10 + a1[c+1] * w11 + a1[c+2] * w12
                        + a2[c]   * w20 + a2[c+1] * w21 + a2[c+2] * w22;
                hrow[c] = s > 0.0f ? s : 0.0f;   // ReLU
            }
        }
    }
    __syncthreads();

    // ---- Phase 3: stage W^T into LDS (reuse xs): Wt[n][k], zero-pad n>=10 --
    {
        float* Wt = xs;
        for (int i = tid; i < 16 * FEAT; i += TPB) {
            const int n = i / FEAT;
            const int k = i - n * FEAT;
            Wt[n * HSTR + k] = (n < 10) ? W[k * 10 + n] : 0.0f;
        }
    }
    __syncthreads();

    // ---- Phase 4: WMMA fp32 16x16x4; K = 676 = 169 chunks over 4 waves -----
    {
        const int kchunks = FEAT / 4;            // 169 (exact, no K padding)
        const int per = (kchunks + 3) / 4;       // 43 -> waves get 43/43/43/40
        const int c0  = wv * per;                // scalar (SGPR) bounds
        int       c1  = c0 + per;
        if (c1 > kchunks) c1 = kchunks;

        const int grp = lane >> 4;               // 0: K pair k0..k0+1, 1: k0+2..+3
        const int mn  = lane & 15;               // A: row m; B: col n
        const float* ha = hs + mn * HSTR + 2 * grp;   // A fragment base
        const float* hb = xs + mn * HSTR + 2 * grp;   // B fragment base (Wt)

        v8f acc = {0.f, 0.f, 0.f, 0.f, 0.f, 0.f, 0.f, 0.f};

        int ch = c0;
        // 4x unrolled: 8 ds_load_b64 issued ahead of 4 chained WMMAs
        for (; ch + 4 <= c1; ch += 4) {
            v2f a0 = *(const v2f*)(ha + 4 * (ch + 0));
            v2f b0 = *(const v2f*)(hb + 4 * (ch + 0));
            v2f a1 = *(const v2f*)(ha + 4 * (ch + 1));
            v2f b1 = *(const v2f*)(hb + 4 * (ch + 1));
            v2f a2 = *(const v2f*)(ha + 4 * (ch + 2));
            v2f b2 = *(const v2f*)(hb + 4 * (ch + 2));
            v2f a3 = *(const v2f*)(ha + 4 * (ch + 3));
            v2f b3 = *(const v2f*)(hb + 4 * (ch + 3));
            acc = __builtin_amdgcn_wmma_f32_16x16x4_f32(false, a0, false, b0,
                                                        (short)0, acc, false, false);
            acc = __builtin_amdgcn_wmma_f32_16x16x4_f32(false, a1, false, b1,
                                                        (short)0, acc, false, false);
            acc = __builtin_amdgcn_wmma_f32_16x16x4_f32(false, a2, false, b2,
                                                        (short)0, acc, false, false);
            acc = __builtin_amdgcn_wmma_f32_16x16x4_f32(false, a3, false, b3,
                                                        (short)0, acc, false, false);
        }
        for (; ch < c1; ++ch) {                  // remainder (<= 3 chunks)
            v2f a = *(const v2f*)(ha + 4 * ch);
            v2f b = *(const v2f*)(hb + 4 * ch);
            acc = __builtin_amdgcn_wmma_f32_16x16x4_f32(false, a, false, b,
                                                        (short)0, acc, false, false);
        }

        #pragma unroll
        for (int r = 0; r < 8; ++r)
            red[wv * 256 + r * 32 + lane] = acc[r];
    }
    __syncthreads();

    // ---- Phase 5: wave 0 reduces 4 K-partials, adds bias, stores 16x10 -----
    if (wv == 0) {
        const int grp = lane >> 4;
        const int n   = lane & 15;
        const float bn = (n < 10) ? bias[n] : 0.0f;
        #pragma unroll
        for (int r = 0; r < 8; ++r) {
            float v = red[0 * 256 + r * 32 + lane]
                    + red[1 * 256 + r * 32 + lane]
                    + red[2 * 256 + r * 32 + lane]
                    + red[3 * 256 + r * 32 + lane];
            const int m = r + 8 * grp;           // C/D layout: VGPR r, lane group
            if (n < 10)
                out[(size_t)(imgbase + m) * 10 + n] = v + bn;
        }
    }
}

extern "C" void kernel_launch(void* const* d_in, const int* in_sizes, int n_in,
                              void* d_out, int out_size, void* d_ws, size_t ws_size,
                              hipStream_t stream) {
    const float* x  = (const float*)d_in[0];   // [B, 784] fp32
    const float* cw = (const float*)d_in[1];   // [3, 3]   fp32
    const float* W  = (const float*)d_in[2];   // [676,10] fp32
    const float* b  = (const float*)d_in[3];   // [10]     fp32
    float* out = (float*)d_out;                // [B, 10]  fp32

    const int B = in_sizes[0] / (IMGDIM * IMGDIM);   // 32768
    const int grid = B / IPB;                        // 2048 blocks

    fused_conv_relu_gemm<<<grid, TPB, 0, stream>>>(x, cw, W, b, out);
}